// SparseConvWithSkips_21010980012297
// MI455X (gfx1250) — compile-verified
//
#include <hip/hip_runtime.h>

// CDNA5 / gfx1250, wave32. f32 WMMA path: V_WMMA_F32_16X16X4_F32.
typedef __attribute__((ext_vector_type(2))) float v2f;
typedef __attribute__((ext_vector_type(8))) float v8f;

#define NV      1000000          // voxels per grid
#define KT      27               // taps
#define CH      8                // channels
#define STEPS   (KT * CH / 4)    // 54 WMMA K-steps per tile
#define BPL     (STEPS * 64)     // packed-B floats per layer (3456)
#define NTILES  (NV / 16)        // 62500
#define NPAIRS  (NTILES / 2)     // 31250 tile-pairs (two tiles per wave)
#define WPB     8                // waves per block (256 threads)

// ---------------------------------------------------------------------------
// Pack w[k][i][o] (27x8x8) into the V_WMMA_F32_16X16X4_F32 B-fragment lane
// layout for all 6 layers. Flat layout: bp[layer][step][lane][j], j in {0,1}.
// B rows of a K=4 step: vgpr j, half-wave hi -> row r = j + 2*hi, ki = 4*s + r.
// Column n = lane&15; columns 8..15 are zero (unused outputs).
// ---------------------------------------------------------------------------
__global__ void pack_weights_kernel(const float* __restrict__ w0,
                                    const float* __restrict__ w1,
                                    const float* __restrict__ w2,
                                    const float* __restrict__ w3,
                                    const float* __restrict__ w4,
                                    const float* __restrict__ w5,
                                    float* __restrict__ bp) {
    int idx = blockIdx.x * blockDim.x + threadIdx.x;
    if (idx >= 6 * BPL) return;
    int layer = idx / BPL;
    int r0    = idx % BPL;
    int s     = r0 >> 6;         // step 0..53
    int t     = r0 & 63;
    int lane  = t >> 1;
    int j     = t & 1;
    int n     = lane & 15;
    int hi    = lane >> 4;
    int ki    = 4 * s + j + 2 * hi;   // 0..215 ; w flat index = ki*CH + o
    const float* w = (layer == 0) ? w0 : (layer == 1) ? w1 : (layer == 2) ? w2
                   : (layer == 3) ? w3 : (layer == 4) ? w4 : w5;
    float v = 0.0f;
    if (n < CH) v = w[ki * CH + n];
    bp[idx] = v;
}

// ---------------------------------------------------------------------------
// One conv layer: out[n,o] = sum_{k,i} x[nbr[n,k], i] * w[k,i,o]
// One wave processes TWO 16-voxel tiles with independent accumulators so
// consecutive v_wmma ops have no D->C dependency (keeps the XDL pipe full)
// and each B-fragment LDS load is reused by both tiles.
// A-frag: lane L (m = L&15, hi = L>>4) holds x[row][(s&1)*4 + 2*hi + {0,1}],
//   row = nbr[base+m][k]  -> two b64 gathers per tap per tile.
// ---------------------------------------------------------------------------
template <bool RELU, bool RESID>
__global__ __launch_bounds__(256)
void conv_layer_kernel(const float* __restrict__ x,
                       const int*   __restrict__ nbr,
                       const float* __restrict__ bp,     // this layer's pack
                       const float* __restrict__ resid,  // feat for residual
                       float*       __restrict__ out,
                       int nPairs) {
    __shared__ float sB[BPL];
    for (int t = threadIdx.x; t < BPL; t += 256) sB[t] = bp[t];
    __syncthreads();

    const int lane = threadIdx.x & 31;
    const int wave = threadIdx.x >> 5;
    const int pair = blockIdx.x * WPB + wave;
    if (pair >= nPairs) return;              // wave-uniform exit

    const int base0 = pair * 32;             // first 16-voxel tile
    const int base1 = base0 + 16;            // second 16-voxel tile
    const int m16   = lane & 15;
    const int hi    = lane >> 4;             // half-wave select

    const int* nrow0 = nbr + (base0 + m16) * KT;
    const int* nrow1 = nbr + (base1 + m16) * KT;

    v8f acc0 = {0.f, 0.f, 0.f, 0.f, 0.f, 0.f, 0.f, 0.f};
    v8f acc1 = {0.f, 0.f, 0.f, 0.f, 0.f, 0.f, 0.f, 0.f};

#pragma unroll 3
    for (int k = 0; k < KT; ++k) {
        const int r0 = nrow0[k];
        const int r1 = nrow1[k];
        const float* xr0 = x + (long)r0 * CH + hi * 2;
        const float* xr1 = x + (long)r1 * CH + hi * 2;
        v2f a00 = *(const v2f*)(xr0);        // tile0, channels {c0,c0+1}
        v2f a01 = *(const v2f*)(xr0 + 4);    // tile0, channels {c0+4,c0+5}
        v2f a10 = *(const v2f*)(xr1);        // tile1
        v2f a11 = *(const v2f*)(xr1 + 4);
        const float* bq = sB + (2 * k) * 64 + lane * 2;
        v2f b0 = *(const v2f*)(bq);          // step 2k   B-frag (shared)
        v2f b1 = *(const v2f*)(bq + 64);     // step 2k+1 B-frag (shared)
        // Interleave the two independent accumulation chains.
        acc0 = __builtin_amdgcn_wmma_f32_16x16x4_f32(false, a00, false, b0,
                                                     (short)0, acc0, false, false);
        acc1 = __builtin_amdgcn_wmma_f32_16x16x4_f32(false, a10, false, b0,
                                                     (short)0, acc1, false, false);
        acc0 = __builtin_amdgcn_wmma_f32_16x16x4_f32(false, a01, false, b1,
                                                     (short)0, acc0, false, false);
        acc1 = __builtin_amdgcn_wmma_f32_16x16x4_f32(false, a11, false, b1,
                                                     (short)0, acc1, false, false);
    }

    // D layout: vgpr j, lane L -> voxel base + j + 8*hi, channel n = L&15.
    // Only channels 0..7 are real; predicate AFTER all WMMAs (EXEC all-1s above).
    if (m16 < CH) {
        const int n  = m16;
        const int rb0 = base0 + hi * 8;
        const int rb1 = base1 + hi * 8;
#pragma unroll
        for (int j = 0; j < 8; ++j) {
            float v0 = acc0[j];
            float v1 = acc1[j];
            if (RELU) { v0 = fmaxf(v0, 0.0f); v1 = fmaxf(v1, 0.0f); }
            const int o0 = (rb0 + j) * CH + n;
            const int o1 = (rb1 + j) * CH + n;
            if (RESID) { v0 += resid[o0]; v1 += resid[o1]; }
            out[o0] = v0;
            out[o1] = v1;
        }
    }
}

// ---------------------------------------------------------------------------
// Launch: pack weights once, then 3 layers per grid (double-buffered in ws).
// ws layout: t0 [NV*CH f32] | t1 [NV*CH f32] | bpack [6*BPL f32]  (~64.1 MB)
// ---------------------------------------------------------------------------
extern "C" void kernel_launch(void* const* d_in, const int* in_sizes, int n_in,
                              void* d_out, int out_size, void* d_ws, size_t ws_size,
                              hipStream_t stream) {
    const float* feat0 = (const float*)d_in[0];
    const float* feat1 = (const float*)d_in[1];
    const float* w00   = (const float*)d_in[2];
    const float* w01   = (const float*)d_in[3];
    const float* w02   = (const float*)d_in[4];
    const float* w10   = (const float*)d_in[5];
    const float* w11   = (const float*)d_in[6];
    const float* w12   = (const float*)d_in[7];
    const int*   nbr0  = (const int*)d_in[8];
    const int*   nbr1  = (const int*)d_in[9];
    float*       out   = (float*)d_out;          // [2][NV][CH]

    float* t0    = (float*)d_ws;
    float* t1    = t0 + (size_t)NV * CH;
    float* bpack = t1 + (size_t)NV * CH;

    // Pack all 6 layers' weights into WMMA B-fragment layout.
    {
        int total  = 6 * BPL;
        int blocks = (total + 255) / 256;
        pack_weights_kernel<<<blocks, 256, 0, stream>>>(w00, w01, w02,
                                                        w10, w11, w12, bpack);
    }

    const int blocks = (NPAIRS + WPB - 1) / WPB;

    // Grid 0
    conv_layer_kernel<true,  false><<<blocks, 256, 0, stream>>>(feat0, nbr0, bpack + 0 * BPL, nullptr, t0, NPAIRS);
    conv_layer_kernel<true,  false><<<blocks, 256, 0, stream>>>(t0,    nbr0, bpack + 1 * BPL, nullptr, t1, NPAIRS);
    conv_layer_kernel<false, true ><<<blocks, 256, 0, stream>>>(t1,    nbr0, bpack + 2 * BPL, feat0,  out, NPAIRS);

    // Grid 1
    conv_layer_kernel<true,  false><<<blocks, 256, 0, stream>>>(feat1, nbr1, bpack + 3 * BPL, nullptr, t0, NPAIRS);
    conv_layer_kernel<true,  false><<<blocks, 256, 0, stream>>>(t0,    nbr1, bpack + 4 * BPL, nullptr, t1, NPAIRS);
    conv_layer_kernel<false, true ><<<blocks, 256, 0, stream>>>(t1,    nbr1, bpack + 5 * BPL, feat1,
                                                                out + (size_t)NV * CH, NPAIRS);
}